// Occupancy_Loss_25683904430791
// MI455X (gfx1250) — compile-verified
//
#include <hip/hip_runtime.h>

#define HH 512
#define WW 1024
#define GRIDN 512
#define NB 16
#define NBINS (GRIDN * GRIDN)          // 262144
#define NPIX (HH * WW)                 // 524288
#define CNT_BYTES (2ull * NB * NBINS * 4ull)   // 33,554,432
#define PI_F 3.14159265358979323846f

typedef float v2f __attribute__((ext_vector_type(2)));
typedef float v8f __attribute__((ext_vector_type(8)));

// Exact f32 sum of all 32 lanes' p using V_WMMA_F32_16X16X4_F32 twice.
// A-layout (ISA 7.12.2): lanes0-15 VGPR0 -> A[m][0], lanes16-31 VGPR0 -> A[m][2].
// With B = ones(4x16): D[m][n] = rowsumA(m) = p[m] + p[m+16].
// Fold D's 8 VGPRs per lane, feed back as A: rowsum2(m) = total. D2[0] = total in all lanes.
__device__ __forceinline__ float wave_sum_wmma(float p) {
    v2f a;  a[0] = p;    a[1] = 0.0f;
    v2f on; on[0] = 1.0f; on[1] = 1.0f;
    v8f cz = {0.0f, 0.0f, 0.0f, 0.0f, 0.0f, 0.0f, 0.0f, 0.0f};
    v8f d = __builtin_amdgcn_wmma_f32_16x16x4_f32(false, a, false, on, (short)0, cz,
                                                  false, false);
    float s = d[0] + d[1] + d[2] + d[3] + d[4] + d[5] + d[6] + d[7];
    v2f a2; a2[0] = s; a2[1] = 0.0f;
    v8f d2 = __builtin_amdgcn_wmma_f32_16x16x4_f32(false, a2, false, on, (short)0, cz,
                                                   false, false);
    return d2[0];
}

__device__ __forceinline__ int bin_of(float depth, float dir) {
    // mirror reference op order: ((d*dir*1000 + 10000) / 20000) * 512, RNE round, clip
    float mm = depth * dir * 1000.0f;
    float c  = (mm + 10000.0f) / 20000.0f * (float)GRIDN;
    float r  = rintf(c);                       // RNE == jnp.round
    r = fminf(fmaxf(r, 0.0f), (float)(GRIDN - 1));
    return (int)r;
}

// One thread per (batch, pixel); scatter both pred and gt counts.
__global__ void occ_hist_kernel(const float* __restrict__ pred,
                                const float* __restrict__ gt,
                                unsigned* __restrict__ cnt, int plane) {
    unsigned tid = blockIdx.x * blockDim.x + threadIdx.x;   // b*NPIX + p
    unsigned b = tid / NPIX;
    unsigned p = tid - b * NPIX;
    unsigned i = p / WW;
    unsigned j = p - i * WW;

    float theta = -PI_F * ((float)i / (float)(HH - 1) - 0.5f);
    float phi   =  PI_F * (2.0f * (float)j / (float)(WW - 1) - 1.0f);
    float st, ct, sp, cp;
    sincosf(theta, &st, &ct);
    sincosf(phi,   &sp, &cp);

    float dx, dy;
    if (plane == 0)      { dx = cp * ct; dy = sp * ct; }   // horizontal
    else if (plane == 1) { dx = cp * ct; dy = -st;     }   // vertical
    else                 { dx = sp * ct; dy = -st;     }   // lateral

    float dp = pred[tid];
    float dg = gt[tid];

    int fx = bin_of(dp, dx), fy = bin_of(dp, dy);
    atomicAdd(&cnt[(unsigned)b * NBINS + (unsigned)fy * GRIDN + fx], 1u);
    fx = bin_of(dg, dx); fy = bin_of(dg, dy);
    atomicAdd(&cnt[(unsigned)(NB + b) * NBINS + (unsigned)fy * GRIDN + fx], 1u);
}

// 1024 blocks per histogram, 32 histograms. max of min(count,100).
__global__ void occ_hmax_kernel(const unsigned* __restrict__ cnt,
                                unsigned* __restrict__ hmax) {
    unsigned h = blockIdx.x >> 10;
    unsigned k = ((blockIdx.x & 1023u) << 8) + threadIdx.x;
    unsigned v = cnt[h * NBINS + k];
    v = v > 100u ? 100u : v;
    #pragma unroll
    for (int m = 16; m >= 1; m >>= 1) {
        unsigned o = (unsigned)__shfl_xor((int)v, m, 32);
        v = v > o ? v : o;
    }
    __shared__ unsigned ls[8];
    unsigned lane = threadIdx.x & 31u, wave = threadIdx.x >> 5;
    if (lane == 0) ls[wave] = v;
    __syncthreads();
    if (threadIdx.x == 0) {
        unsigned bm = ls[0];
        #pragma unroll
        for (int w = 1; w < 8; ++w) bm = bm > ls[w] ? bm : ls[w];
        atomicMax(&hmax[h], bm);
    }
}

__device__ __forceinline__ float occ_diff(const unsigned* cnt, const unsigned* hmax,
                                          unsigned b, unsigned k) {
    unsigned cp = cnt[b * NBINS + k];            cp = cp > 100u ? 100u : cp;
    unsigned cg = cnt[(NB + b) * NBINS + k];     cg = cg > 100u ? 100u : cg;
    float fp = (float)cp / (float)hmax[b];
    float fg = (float)cg / (float)hmax[NB + b];
    return fabsf(fp - fg);
}

// Global (all-sphere) max |pred_norm - gt_norm| via integer atomicMax on float bits (>=0).
__global__ void occ_dmax_kernel(const unsigned* __restrict__ cnt,
                                const unsigned* __restrict__ hmax,
                                unsigned* __restrict__ dmax) {
    unsigned b = blockIdx.x >> 10;
    unsigned k = ((blockIdx.x & 1023u) << 8) + threadIdx.x;
    float diff = occ_diff(cnt, hmax, b, k);
    #pragma unroll
    for (int m = 16; m >= 1; m >>= 1) {
        float o = __shfl_xor(diff, m, 32);
        diff = fmaxf(diff, o);
    }
    __shared__ float ls[8];
    unsigned lane = threadIdx.x & 31u, wave = threadIdx.x >> 5;
    if (lane == 0) ls[wave] = diff;
    __syncthreads();
    if (threadIdx.x == 0) {
        float bm = ls[0];
        #pragma unroll
        for (int w = 1; w < 8; ++w) bm = fmaxf(bm, ls[w]);
        atomicMax(dmax, __float_as_uint(bm));
    }
}

// berHu per element, WMMA wave-sum, one deterministic block sum per block.
__global__ void occ_loss_kernel(const unsigned* __restrict__ cnt,
                                const unsigned* __restrict__ hmax,
                                const unsigned* __restrict__ dmax,
                                float* __restrict__ bsum) {
    float c = 0.2f * __uint_as_float(*dmax);
    unsigned b = blockIdx.x >> 10;
    unsigned k = ((blockIdx.x & 1023u) << 8) + threadIdx.x;
    float diff = occ_diff(cnt, hmax, b, k);
    float l = (diff <= c) ? diff : (diff * diff + c * c) / (2.0f * c);
    float wtot = wave_sum_wmma(l);               // EXEC all-ones here: no divergence
    __shared__ float ls[8];
    unsigned lane = threadIdx.x & 31u, wave = threadIdx.x >> 5;
    if (lane == 0) ls[wave] = wtot;
    __syncthreads();
    if (threadIdx.x == 0) {
        float s = 0.0f;
        #pragma unroll
        for (int w = 0; w < 8; ++w) s += ls[w];
        bsum[blockIdx.x] = s;
    }
}

// Single block: deterministic final sum of 16384 block sums; out += mean.
__global__ void occ_final_kernel(const float* __restrict__ bsum,
                                 float* __restrict__ out, int n) {
    float p = 0.0f;
    for (int i = threadIdx.x; i < n; i += 256) p += bsum[i];
    float wtot = wave_sum_wmma(p);
    __shared__ float ls[8];
    unsigned lane = threadIdx.x & 31u, wave = threadIdx.x >> 5;
    if (lane == 0) ls[wave] = wtot;
    __syncthreads();
    if (threadIdx.x == 0) {
        float s = 0.0f;
        #pragma unroll
        for (int w = 0; w < 8; ++w) s += ls[w];
        out[0] += s * (1.0f / (float)(NB * NBINS));
    }
}

extern "C" void kernel_launch(void* const* d_in, const int* in_sizes, int n_in,
                              void* d_out, int out_size, void* d_ws, size_t ws_size,
                              hipStream_t stream) {
    (void)in_sizes; (void)n_in; (void)out_size; (void)ws_size;
    const float* pred = (const float*)d_in[0];
    const float* gt   = (const float*)d_in[1];
    unsigned char* ws = (unsigned char*)d_ws;

    unsigned* cnt  = (unsigned*)ws;                          // 33,554,432 B
    unsigned* hmax = (unsigned*)(ws + CNT_BYTES);            // 32 u32
    unsigned* dmax = (unsigned*)(ws + CNT_BYTES + 128);      // 1 u32 (float bits)
    float*    bsum = (float*)(ws + CNT_BYTES + 256);         // 16384 floats
    float*    out  = (float*)d_out;

    hipMemsetAsync(d_out, 0, sizeof(float), stream);

    const int histBlocks = (NB * NPIX) / 256;    // 32768
    const int hmaxBlocks = 32 * 1024;            // 32 hists * 1024
    const int elemBlocks = 16 * 1024;            // 16 batches * 1024

    for (int s = 0; s < 3; ++s) {
        // zero counts + hmax + dmax (cnt..dmax contiguous)
        hipMemsetAsync(cnt, 0, CNT_BYTES + 256, stream);
        occ_hist_kernel<<<histBlocks, 256, 0, stream>>>(pred, gt, cnt, s);
        occ_hmax_kernel<<<hmaxBlocks, 256, 0, stream>>>(cnt, hmax);
        occ_dmax_kernel<<<elemBlocks, 256, 0, stream>>>(cnt, hmax, dmax);
        occ_loss_kernel<<<elemBlocks, 256, 0, stream>>>(cnt, hmax, dmax, bsum);
        occ_final_kernel<<<1, 256, 0, stream>>>(bsum, out, elemBlocks);
    }
}